// TemproalSelfMatrix_77077483094180
// MI455X (gfx1250) — compile-verified
//
#include <hip/hip_runtime.h>

typedef float v2f __attribute__((ext_vector_type(2)));
typedef float v8f __attribute__((ext_vector_type(8)));

#define NB   8
#define S    2048
#define E    64
#define TEMP 13.544f
#define ROW_PAD 2052   // 2048 + 4 floats: rows differ by 4 banks, halves by 32 banks

__global__ __launch_bounds__(256)
void dist_softmax_wmma(const float* __restrict__ x, float* __restrict__ out) {
    extern __shared__ float lds[];
    float* sqs  = lds;        // [16]  row sums-of-squares for this s-block
    float* dist = lds + 16;   // [16][ROW_PAD]

    const int tid    = threadIdx.x;
    const int wave   = tid >> 5;
    const int lane   = tid & 31;
    const int laneLo = lane & 15;
    const int hi     = lane >> 4;          // 0: K%4 in {0,1}; 1: K%4 in {2,3}

    const int n    = blockIdx.x >> 7;      // batch
    const int s0   = (blockIdx.x & 127) * 16;

    const float* xb = x + (size_t)n * S * E;

    // ||x_s||^2 for the 16 A rows (threads 0..15)
    if (tid < 16) {
        const float* r = xb + (size_t)(s0 + tid) * E;
        float acc = 0.f;
        #pragma unroll
        for (int k = 0; k < E; ++k) acc += r[k] * r[k];
        sqs[tid] = acc;
    }

    // A fragments: M = s0 + laneLo, 16 K-steps of 4, split by lane half
    const float* arow = xb + (size_t)(s0 + laneLo) * E + 2 * hi;
    v2f a[16];
    #pragma unroll
    for (int k = 0; k < 16; ++k)
        a[k] = *(const v2f*)(arow + 4 * k);

    __syncthreads();

    // C/D layout: lane holds rows M = v + 8*hi  (v = accumulator VGPR index)
    float sqsv[8];
    #pragma unroll
    for (int v = 0; v < 8; ++v) sqsv[v] = sqs[v + 8 * hi];

    // each wave: 16 consecutive g-tiles => columns [wave*256, wave*256+256)
    for (int i = 0; i < 16; ++i) {
        const int g0 = wave * 256 + i * 16;
        const float* brow = xb + (size_t)(g0 + laneLo) * E + 2 * hi;
        v2f bf[16];
        float psq = 0.f;
        #pragma unroll
        for (int k = 0; k < 16; ++k) {
            bf[k] = *(const v2f*)(brow + 4 * k);
            psq += bf[k].x * bf[k].x + bf[k].y * bf[k].y;
        }
        // lane N and lane N+16 hold the two K-halves of column N
        const float sqg = psq + __shfl_xor(psq, 16, 32);

        v8f c = {};
        #pragma unroll
        for (int k = 0; k < 16; ++k)
            c = __builtin_amdgcn_wmma_f32_16x16x4_f32(
                    /*neg_a=*/false, a[k], /*neg_b=*/false, bf[k],
                    /*c_mod=*/(short)0, c, /*reuse_a=*/false, /*reuse_b=*/false);

        #pragma unroll
        for (int v = 0; v < 8; ++v) {
            float d = fmaf(-2.0f, c[v], sqsv[v] + sqg);   // ||s||^2+||g||^2-2<s,g>
            d = fmaxf(d, 0.0f);
            dist[(v + 8 * hi) * ROW_PAD + g0 + laneLo] = d;
        }
    }

    __syncthreads();

    // softmax(-dist/T): row max of -d/T == -min(d)/T. Each wave: 2 rows.
    const float invT = 1.0f / TEMP;
    #pragma unroll
    for (int rr = 0; rr < 2; ++rr) {
        const int r = wave * 2 + rr;
        float* row = dist + r * ROW_PAD;

        float m = 3.4e38f;
        for (int cI = lane; cI < S; cI += 32) m = fminf(m, row[cI]);
        #pragma unroll
        for (int o = 16; o > 0; o >>= 1) m = fminf(m, __shfl_xor(m, o, 32));

        float s = 0.f;
        for (int cI = lane; cI < S; cI += 32) {
            float e = __expf((m - row[cI]) * invT);
            row[cI] = e;
            s += e;
        }
        #pragma unroll
        for (int o = 16; o > 0; o >>= 1) s += __shfl_xor(s, o, 32);
        const float inv_s = 1.0f / s;

        float* orow = out + (size_t)n * S * S + (size_t)(s0 + r) * S;
        #pragma unroll
        for (int j = 0; j < 16; ++j) {          // 32 lanes * 16 * float4 = 2048
            const int c4 = (lane + j * 32) * 4;
            float4 v4 = *(float4*)(row + c4);
            v4.x *= inv_s; v4.y *= inv_s; v4.z *= inv_s; v4.w *= inv_s;
            *(float4*)(orow + c4) = v4;          // single wide store per element
        }
    }
}

extern "C" void kernel_launch(void* const* d_in, const int* in_sizes, int n_in,
                              void* d_out, int out_size, void* d_ws, size_t ws_size,
                              hipStream_t stream) {
    (void)in_sizes; (void)n_in; (void)out_size; (void)d_ws; (void)ws_size;
    const float* x = (const float*)d_in[0];
    float* out = (float*)d_out;
    const size_t ldsBytes = (size_t)(16 + 16 * ROW_PAD) * sizeof(float); // ~128.3 KB
    dim3 grid(NB * (S / 16));   // 1024 workgroups: (batch, s-block)
    dim3 block(256);            // 8 waves (wave32)
    dist_softmax_wmma<<<grid, block, ldsBytes, stream>>>(x, out);
}